// ConsistencyLoss_15401752723721
// MI455X (gfx1250) — compile-verified
//
#include <hip/hip_runtime.h>

// ---------------------------------------------------------------------------
// ConsistencyLoss for MI455X (gfx1250): masked cosine-similarity reduction
// via f32 WMMA (V_WMMA_F32_16X16X4_F32), wave32.
//
//   out = -( sum_b sum_{ij in mask} [cos(y_bi, zp_bj) + cos(z_bi, yp_bj)] )
//         / (n_pairs * B)
//
// B=64, C=256, N=H*W=784.  A-side pixel vectors are pre-normalized and staged
// in LDS; B-side is streamed raw from L2 and its 1/||b_j|| norm is applied as
// a per-column post-scale (masked sum is linear).  C is processed in two
// halves of 128 channels (also linear) to keep the 64-row A tile in <64KB LDS.
// ---------------------------------------------------------------------------

typedef __attribute__((ext_vector_type(2))) float v2f;
typedef __attribute__((ext_vector_type(8))) float v8f;

#define B_DIM   64
#define C_DIM   256
#define C_HALF  128
#define N_PIX   784          // 28*28 = 49 tiles of 16
#define THRESH  31.5f
#define A_STR   132          // 128 + 4 pad: stride mod 64 == 4 -> conflict-free b64 frag loads

// ---------------------------------------------------------------------------
// ws layout (floats): [0] total accumulator, [1] pair count (as int),
//                     [4 ..] rnorm[4][B][N]  (tensor order: y, yp, z, zp)
// ---------------------------------------------------------------------------

__global__ void cl_init_kernel(float* total, int* cnt) {
    total[0] = 0.0f;
    cnt[0]   = 0;
}

// One thread per (tensor, b, pixel); consecutive threads -> consecutive pixels
// so the per-channel loop (stride N_PIX) is coalesced across the wave.
__global__ void cl_norms_kernel(const float* __restrict__ y,
                                const float* __restrict__ yp,
                                const float* __restrict__ z,
                                const float* __restrict__ zp,
                                float* __restrict__ rn) {
    const int per = B_DIM * N_PIX;
    int idx = blockIdx.x * blockDim.x + threadIdx.x;      // [tensor][b][p]
    int t   = idx / per;
    int r   = idx % per;
    const float* src = (t == 0) ? y : (t == 1) ? yp : (t == 2) ? z : zp;
    const float* base = src + (size_t)(r / N_PIX) * C_DIM * N_PIX + (r % N_PIX);
    float ss = 0.0f;
#pragma unroll 8
    for (int c = 0; c < C_DIM; ++c) {
        float v = base[(size_t)c * N_PIX];
        ss += v * v;
    }
    rn[idx] = rsqrtf(fmaxf(ss, 1e-24f));   // norms ~16 for these inputs; EPS never binds
}

__global__ void cl_count_kernel(const float* __restrict__ dist, int* __restrict__ cnt) {
    __shared__ int red[256];
    int idx = blockIdx.x * 256 + threadIdx.x;             // 784*784 = 2401 * 256 exactly
    red[threadIdx.x] = (dist[idx] < THRESH) ? 1 : 0;
    __syncthreads();
    for (int s = 128; s > 0; s >>= 1) {
        if (threadIdx.x < s) red[threadIdx.x] += red[threadIdx.x + s];
        __syncthreads();
    }
    if (threadIdx.x == 0) atomicAdd(cnt, red[0]);
}

// ---------------------------------------------------------------------------
// Main kernel: grid (13 i-groups of 64 rows, 64 batches, 2 pairs), 128 thr.
// Wave w: i-strip (w>>1) of 32 rows, j-tiles of parity (w&1).
// Per j-tile: 32x16x128 via 2 chained WMMA/k-step sharing one B fragment
// loaded straight from global (L2-resident), then masked + rnorm-B scaled.
// ---------------------------------------------------------------------------
__global__ void __launch_bounds__(128)
cl_pixwise_kernel(const float* __restrict__ y,  const float* __restrict__ yp,
                  const float* __restrict__ z,  const float* __restrict__ zp,
                  const float* __restrict__ dist,
                  const float* __restrict__ rn,
                  float* __restrict__ total) {
    __shared__ float As[64 * A_STR];     // 64 pixels x 128 channels (half of C)

    const int tid    = threadIdx.x;
    const int lane   = tid & 31;
    const int wave   = tid >> 5;
    const int istrip = wave >> 1;        // 0/1 -> rows [32*istrip, +32) of the block
    const int jpar   = wave & 1;         // j-tile parity
    const int b      = blockIdx.y;
    const int pair   = blockIdx.z;       // 0: (y, zp)   1: (z, yp)
    const int i0     = blockIdx.x * 64;
    const int per    = B_DIM * N_PIX;

    const float* Ag  = pair ? z  : y;
    const float* Bg  = pair ? yp : zp;
    const float* rnA = rn + (pair ? 2 : 0) * per + b * N_PIX;
    const float* rnB = rn + (pair ? 1 : 3) * per + b * N_PIX;
    const float* Abase = Ag + (size_t)b * C_DIM * N_PIX;
    const float* Bbase = Bg + (size_t)b * C_DIM * N_PIX;

    const int  m  = lane & 15;           // A row within tile / D column
    const int  h  = lane >> 4;           // K-half selector (ISA 16x4 f32 layouts)
    const int  ib = istrip * 32;
    const bool wvalid = (i0 + ib) < N_PIX;

    // staging-invariant pixel for this thread
    const int   sp = tid & 63;
    const bool  pv = (i0 + sp) < N_PIX;  // last group: only 16 of 64 rows valid
    const float sA = pv ? rnA[i0 + sp] : 0.0f;
    const float* gA = Abase + (i0 + sp);

    float local = 0.0f;

    for (int kh = 0; kh < 2; ++kh) {     // two halves of the channel dimension
        const int cbase = kh * C_HALF;

        // ---- Stage normalized A half-strip: global [c][pix] -> LDS [pix][c]
        for (int idx = tid; idx < 64 * C_HALF; idx += 128) {
            int c = idx >> 6;            // sp = idx & 63 is thread-invariant
            As[sp * A_STR + c] = pv ? gA[(size_t)(cbase + c) * N_PIX] * sA : 0.0f;
        }
        __syncthreads();

        if (wvalid) {
            const float* ap0 = &As[(ib + m) * A_STR + 2 * h];        // rows ib..ib+15
            const float* ap1 = ap0 + 16 * A_STR;                     // rows ib+16..ib+31

            for (int jt = jpar; jt < 49; jt += 2) {
                const int j0 = jt * 16;
                // raw B column stream: B[cbase+2h + k + {0,1}][j0 + m]
                const float* Bp = Bbase + ((size_t)cbase + 2 * h) * N_PIX + j0 + m;

                v8f acc0 = {}, acc1 = {};
#pragma unroll
                for (int k = 0; k < C_HALF; k += 4) {
                    v2f a0 = *(const v2f*)(ap0 + k);                 // ds_load_b64
                    v2f a1 = *(const v2f*)(ap1 + k);
                    v2f bf;
                    bf.x = Bp[(size_t)k * N_PIX];                    // global_load_b32
                    bf.y = Bp[(size_t)(k + 1) * N_PIX];
                    acc0 = __builtin_amdgcn_wmma_f32_16x16x4_f32(
                               false, a0, false, bf, (short)0, acc0, false, false);
                    acc1 = __builtin_amdgcn_wmma_f32_16x16x4_f32(
                               false, a1, false, bf, (short)0, acc1, false, false);
                }

                // D layout: acc[r] = (row r + 8h, col m); scale by 1/||b_j||, mask
                const int gj = j0 + m;
                const float sB = rnB[gj];
                const int gi0 = i0 + ib + 8 * h;
                const float* dp = dist + (size_t)gi0 * N_PIX + gj;
                float t = 0.0f;
#pragma unroll
                for (int r = 0; r < 8; ++r) {
                    if (gi0 + r < N_PIX && dp[(size_t)r * N_PIX] < THRESH)
                        t += acc0[r];
                    if (gi0 + 16 + r < N_PIX && dp[(size_t)(r + 16) * N_PIX] < THRESH)
                        t += acc1[r];
                }
                local += t * sB;
            }
        }
        __syncthreads();                 // As reuse safety before next C-half
    }

    // ---- wave32 reduction + one atomic per wave
    for (int off = 16; off > 0; off >>= 1)
        local += __shfl_xor(local, off, 32);
    if (lane == 0) atomicAdd(total, local);
}

__global__ void cl_finalize_kernel(const float* total, const int* cnt, float* out) {
    out[0] = -total[0] / ((float)cnt[0] * (float)B_DIM);
}

// ---------------------------------------------------------------------------
extern "C" void kernel_launch(void* const* d_in, const int* in_sizes, int n_in,
                              void* d_out, int out_size, void* d_ws, size_t ws_size,
                              hipStream_t stream) {
    const float* y    = (const float*)d_in[0];
    const float* yp   = (const float*)d_in[1];
    const float* z    = (const float*)d_in[2];
    const float* zp   = (const float*)d_in[3];
    const float* dist = (const float*)d_in[4];
    float* out = (float*)d_out;

    float* wsf   = (float*)d_ws;
    float* total = wsf;                  // [0]
    int*   cnt   = (int*)(wsf + 1);      // [1]
    float* rnbuf = wsf + 4;              // 4 * 64 * 784 floats (~786 KB)

    cl_init_kernel<<<1, 1, 0, stream>>>(total, cnt);

    // 4 tensors * 64 * 784 = 200704 threads = 784 blocks of 256
    cl_norms_kernel<<<784, 256, 0, stream>>>(y, yp, z, zp, rnbuf);

    // 784*784 = 2401 blocks of 256
    cl_count_kernel<<<2401, 256, 0, stream>>>(dist, cnt);

    // 13 i-groups (ceil(784/64)) x 64 batches x 2 pairs, 128 threads (4 waves)
    dim3 grid(13, B_DIM, 2);
    cl_pixwise_kernel<<<grid, 128, 0, stream>>>(y, yp, z, zp, dist, rnbuf, total);

    cl_finalize_kernel<<<1, 1, 0, stream>>>(total, cnt, out);
}